// PZOHGNN_41351945126311
// MI455X (gfx1250) — compile-verified
//
#include <hip/hip_runtime.h>
#include <hip/hip_bf16.h>

#define T_   8
#define V_   10000
#define E_   10000
#define NNZ_ 80000
#define H_   128
#define B_   16

typedef __attribute__((ext_vector_type(2))) float    v2f;
typedef __attribute__((ext_vector_type(8))) float    v8f;
typedef __attribute__((ext_vector_type(4))) unsigned u32x4;
typedef __attribute__((ext_vector_type(4))) int      i32x4;
typedef __attribute__((ext_vector_type(8))) int      i32x8;

// ---- monotonic float<->uint encoding for atomicMax on floats ----
__device__ __forceinline__ unsigned enc_f(float f) {
  unsigned u = __float_as_uint(f);
  return (u & 0x80000000u) ? ~u : (u | 0x80000000u);
}
__device__ __forceinline__ float dec_f(unsigned e) {
  return (e & 0x80000000u) ? __uint_as_float(e & 0x7FFFFFFFu) : __uint_as_float(~e);
}

// =====================================================================
// TDM: stage nelem f32 of W (contiguous, row-major) into LDS offset 0.
// D# per CDNA5 ISA ch.8: 1-D tensor, data_size=4B, tile_dim0=nelem.
// Issue from ONE wave only (TDM ignores EXEC; avoids duplicate DMA).
// =====================================================================
__device__ __forceinline__ void tdm_load_w(const float* W, unsigned nelem) {
  unsigned long long ga = (unsigned long long)(const void*)W;
  u32x4 g0;
  g0[0] = 1u;                                                  // count=1, user mode
  g0[1] = 0u;                                                  // lds_addr = 0 (dynamic LDS base)
  g0[2] = (unsigned)(ga & 0xFFFFFFFFull);                      // global_addr[31:0]
  g0[3] = (unsigned)((ga >> 32) & 0x1FFFFFFull) | (2u << 30);  // global_addr[56:32], type=2
  i32x8 g1;
  g1[0] = (int)(2u << 16);                       // workgroup_mask=0, data_size=2 (4B)
  g1[1] = (int)((nelem & 0xFFFFu) << 16);        // tensor_dim0[15:0]
  g1[2] = (int)((nelem >> 16) | (1u << 16));     // tensor_dim0[31:16], tensor_dim1=1
  g1[3] = (int)((nelem & 0xFFFFu) << 16);        // tensor_dim1 hi=0, tile_dim0=nelem
  g1[4] = 1;                                     // tile_dim1=1, tile_dim2=0
  g1[5] = (int)nelem;                            // tensor_dim0_stride[31:0]
  g1[6] = 0;                                     // stride hi, tensor_dim1_stride lo
  g1[7] = 0;
  i32x4 z4 = {0, 0, 0, 0};
#if __clang_major__ >= 23
  i32x8 z8 = {0, 0, 0, 0, 0, 0, 0, 0};
  __builtin_amdgcn_tensor_load_to_lds(g0, g1, z4, z4, z8, 0);
#else
  __builtin_amdgcn_tensor_load_to_lds(g0, g1, z4, z4, 0);
#endif
}

// =====================================================================
// Generic GEMM: C[N,128] = act(A[N,K] @ W[K,128] + bias), K % 4 == 0.
// W staged in LDS by TDM; one wave -> 16 rows x 128 cols, f32 WMMA 16x16x4.
// B fragments come from LDS (ds_load_2addr pairs), 8 WMMAs issue as a burst.
// =====================================================================
__global__ void wmma_linear_kernel(const float* __restrict__ A,
                                   const float* __restrict__ W,
                                   const float* __restrict__ bias,
                                   float* __restrict__ C,
                                   int N, int K, int do_relu) {
  extern __shared__ float sW[];                 // K*128 floats
  if ((threadIdx.x >> 5) == 0) {
    tdm_load_w(W, (unsigned)(K * 128));
    __builtin_amdgcn_s_wait_tensorcnt(0);
  }
  __syncthreads();

  const int lane = threadIdx.x & 31;
  const int wave = threadIdx.x >> 5;
  const int tile = blockIdx.x * (blockDim.x >> 5) + wave;
  const int row0 = tile * 16;
  if (row0 >= N) return;                        // wave-uniform, after the barrier

  v8f acc[8];
#pragma unroll
  for (int c = 0; c < 8; ++c)
#pragma unroll
    for (int j = 0; j < 8; ++j) acc[c][j] = 0.0f;

  const int m     = lane & 15;                  // M (A) / N (B,C) index
  const int khalf = (lane >> 4) << 1;           // lanes 0-15 -> k0, 16-31 -> k0+2
  const float* arow = A + (size_t)(row0 + m) * K + khalf;

  for (int k0 = 0; k0 < K; k0 += 4) {
    const int kk = k0 + khalf;
    v2f a; a.x = arow[k0]; a.y = arow[k0 + 1];
    float bx[8], by[8];
    const float* w0 = sW + kk * 128 + m;
#pragma unroll
    for (int c = 0; c < 8; ++c) {               // pairs 512B apart -> ds_load_2addr
      bx[c] = w0[c * 16];
      by[c] = w0[c * 16 + 128];
    }
#pragma unroll
    for (int c = 0; c < 8; ++c) {               // independent accs: back-to-back WMMA
      v2f b; b.x = bx[c]; b.y = by[c];
      acc[c] = __builtin_amdgcn_wmma_f32_16x16x4_f32(
          false, a, false, b, (short)0, acc[c], false, false);
    }
  }

  const int rbase = row0 + ((lane >> 4) << 3);  // rows 0-7 lanes 0-15, 8-15 lanes 16-31
#pragma unroll
  for (int c = 0; c < 8; ++c) {
    const int col = c * 16 + m;
    const float bv = bias ? bias[col] : 0.0f;
#pragma unroll
    for (int j = 0; j < 8; ++j) {
      float v = acc[c][j] + bv;
      if (do_relu) v = fmaxf(v, 0.0f);
      C[(size_t)(rbase + j) * 128 + col] = v;
    }
  }
}

// =====================================================================
// Fused conv: X3[N,128] += relu( LN(MSG[N,384] @ W[384,128] + b) * g + bln )
// W (192KB) staged in LDS by TDM; LN row stats via 16-lane shfl_xor.
// =====================================================================
__global__ void wmma_conv_kernel(const float* __restrict__ MSG,
                                 const float* __restrict__ W,
                                 const float* __restrict__ bias,
                                 const float* __restrict__ ln_g,
                                 const float* __restrict__ ln_b,
                                 float* __restrict__ X3, int N) {
  extern __shared__ float sW[];                 // 384*128 floats
  if ((threadIdx.x >> 5) == 0) {
    tdm_load_w(W, 384u * 128u);
    __builtin_amdgcn_s_wait_tensorcnt(0);
  }
  __syncthreads();

  const int lane = threadIdx.x & 31;
  const int wave = threadIdx.x >> 5;
  const int tile = blockIdx.x * (blockDim.x >> 5) + wave;
  const int row0 = tile * 16;
  if (row0 >= N) return;

  v8f acc[8];
#pragma unroll
  for (int c = 0; c < 8; ++c)
#pragma unroll
    for (int j = 0; j < 8; ++j) acc[c][j] = 0.0f;

  const int m     = lane & 15;
  const int khalf = (lane >> 4) << 1;
  const float* arow = MSG + (size_t)(row0 + m) * 384 + khalf;

  for (int k0 = 0; k0 < 384; k0 += 4) {
    const int kk = k0 + khalf;
    v2f a; a.x = arow[k0]; a.y = arow[k0 + 1];
    float bx[8], by[8];
    const float* w0 = sW + kk * 128 + m;
#pragma unroll
    for (int c = 0; c < 8; ++c) {
      bx[c] = w0[c * 16];
      by[c] = w0[c * 16 + 128];
    }
#pragma unroll
    for (int c = 0; c < 8; ++c) {
      v2f b; b.x = bx[c]; b.y = by[c];
      acc[c] = __builtin_amdgcn_wmma_f32_16x16x4_f32(
          false, a, false, b, (short)0, acc[c], false, false);
    }
  }

  // add bias before LN statistics (y = msg@W + b)
#pragma unroll
  for (int c = 0; c < 8; ++c) {
    const float bv = bias[c * 16 + m];
#pragma unroll
    for (int j = 0; j < 8; ++j) acc[c][j] += bv;
  }

  // per-row mean / var: row (row0 + 8*half + j) lives in one 16-lane half
  float mu[8], rinv[8];
#pragma unroll
  for (int j = 0; j < 8; ++j) {
    float s = 0.0f, s2 = 0.0f;
#pragma unroll
    for (int c = 0; c < 8; ++c) { float v = acc[c][j]; s += v; s2 += v * v; }
#pragma unroll
    for (int off = 1; off < 16; off <<= 1) {
      s  += __shfl_xor(s,  off, 32);
      s2 += __shfl_xor(s2, off, 32);
    }
    const float mean = s * (1.0f / 128.0f);
    const float var  = s2 * (1.0f / 128.0f) - mean * mean;
    mu[j]   = mean;
    rinv[j] = rsqrtf(var + 1e-5f);
  }

  const int rbase = row0 + ((lane >> 4) << 3);
#pragma unroll
  for (int c = 0; c < 8; ++c) {
    const int col = c * 16 + m;
    const float g  = ln_g[col];
    const float bb = ln_b[col];
#pragma unroll
    for (int j = 0; j < 8; ++j) {
      float v = (acc[c][j] - mu[j]) * rinv[j] * g + bb;
      v = fmaxf(v, 0.0f);
      const size_t idx = (size_t)(rbase + j) * 128 + col;
      X3[idx] = X3[idx] + v;
    }
  }
}

// =====================================================================
// Message passing scatter: one block per (t, nnz) pair, 128 threads = H.
// MSG row layout: [sum(128) | encoded-max(128) | sumsq(128)]
// =====================================================================
__global__ void scatter_kernel(const float* __restrict__ X,
                               const int* __restrict__ segrow,
                               const int* __restrict__ segcol,
                               float* __restrict__ MSG,
                               int nnz, int SRC, int RDST) {
  const int pair = blockIdx.x;
  const int h = threadIdx.x;
  const int t = pair / nnz;
  const int n = pair - t * nnz;
  const int r = segrow[n];
  const int c = segcol[n];
  const float v = X[((size_t)t * SRC + c) * 128 + h];
  float* base = MSG + ((size_t)t * RDST + r) * 384;
  atomicAdd(base + h, v);
  atomicMax((unsigned*)(base + 128) + h, enc_f(v));
  atomicAdd(base + 256 + h, v * v);
}

__global__ void msg_finalize_kernel(float* __restrict__ MSG,
                                    const float* __restrict__ cnt, int RDST) {
  const size_t idx = (size_t)blockIdx.x * blockDim.x + threadIdx.x;
  const size_t total = (size_t)T_ * RDST * 128;
  if (idx >= total) return;
  const int h = (int)(idx & 127);
  const size_t tr = idx >> 7;
  const int r = (int)(tr % RDST);
  float* base = MSG + tr * 384;
  const float cn = cnt[r];
  const float denom = fmaxf(cn, 1.0f);
  base[h] = base[h] / denom;
  const unsigned mu = ((const unsigned*)base)[128 + h];
  base[128 + h] = (cn > 0.0f) ? dec_f(mu) : 0.0f;
  base[256 + h] = base[256 + h] / denom;
}

__global__ void count_kernel(const int* __restrict__ idxarr,
                             float* __restrict__ cnt, int n) {
  const int i = blockIdx.x * blockDim.x + threadIdx.x;
  if (i < n) atomicAdd(cnt + idxarr[i], 1.0f);
}

// =====================================================================
// Small stage kernels
// =====================================================================
__global__ void label_kernel(const float* __restrict__ xv,
                             const int* __restrict__ tar,
                             const float* __restrict__ W,
                             const float* __restrict__ b,
                             float* __restrict__ lab) {
  const int t = blockIdx.x, h = threadIdx.x;
  const float* x = xv + (size_t)tar[t] * 128;
  float s = b[h];
  for (int k = 0; k < 128; ++k) s += x[k] * W[k * 128 + h];
  lab[t * 128 + h] = fmaxf(s, 0.0f);
}

__global__ void init_xv3_kernel(const float* __restrict__ xv,
                                const float* __restrict__ lab,
                                const int* __restrict__ tar,
                                float* __restrict__ xv3) {
  const int tv = blockIdx.x, h = threadIdx.x;
  const int t = tv / V_, v = tv - t * V_;
  float val = xv[(size_t)v * 128 + h];
  if (v == tar[t]) val = lab[t * 128 + h];
  xv3[(size_t)tv * 128 + h] = val;
}

__global__ void init_xe3_kernel(const float* __restrict__ xe,
                                float* __restrict__ xe3) {
  const int te = blockIdx.x, h = threadIdx.x;
  const int e = te % E_;
  xe3[(size_t)te * 128 + h] = xe[(size_t)e * 128 + h];
}

__global__ void xvf_kernel(const float* __restrict__ xv3,
                           const int* __restrict__ tar,
                           float* __restrict__ xvf) {
  const int i = blockIdx.x, h = threadIdx.x;   // i in [0,64)
  const int t = i / T_, tt = i % T_;
  xvf[(size_t)i * 128 + h] = xv3[((size_t)t * V_ + tar[tt]) * 128 + h];
}

// segmented max/mean pooling: OUT[s] = [max(128) | mean(128)] over entries
// with segrow[j]==s; gather index = segcol[j] (or j itself if segcol==null)
__global__ void pool_seg_kernel(const float* __restrict__ G,
                                const int* __restrict__ segrow,
                                const int* __restrict__ segcol,
                                float* __restrict__ OUT, int nnzp) {
  const int s = blockIdx.x, h = threadIdx.x;
  float mx = -3.402823466e+38f, sm = 0.0f;
  int cnt = 0;
  for (int j = 0; j < nnzp; ++j) {
    if (segrow[j] == s) {
      const int gi = segcol ? segcol[j] : j;
      const float v = G[(size_t)gi * 128 + h];
      mx = fmaxf(mx, v);
      sm += v;
      ++cnt;
    }
  }
  OUT[(size_t)s * 256 + h]       = cnt > 0 ? mx : 0.0f;
  OUT[(size_t)s * 256 + 128 + h] = sm / fmaxf((float)cnt, 1.0f);
}

__global__ void final_kernel(const float* __restrict__ p2h,
                             const float* __restrict__ W1,
                             const float* __restrict__ b1,
                             float* __restrict__ out) {
  __shared__ float red[128];
  const int b = blockIdx.x, h = threadIdx.x;
  red[h] = p2h[(size_t)b * 128 + h] * W1[h];
  __syncthreads();
  for (int off = 64; off > 0; off >>= 1) {
    if (h < off) red[h] += red[h + off];
    __syncthreads();
  }
  if (h == 0) out[b] = red[0] + b1[0];
}

// =====================================================================
extern "C" void kernel_launch(void* const* d_in, const int* in_sizes, int n_in,
                              void* d_out, int out_size, void* d_ws, size_t ws_size,
                              hipStream_t stream) {
  // params flattened in insertion order (indices 0..35), then tensors
  const float* P[36];
  for (int i = 0; i < 36; ++i) P[i] = (const float*)d_in[i];
  const float* xv       = (const float*)d_in[36];
  const float* xe       = (const float*)d_in[37];
  const int*   inci_row = (const int*)d_in[38];
  const int*   inci_col = (const int*)d_in[39];
  const int*   tar      = (const int*)d_in[40];
  const int*   batch    = (const int*)d_in[41];
  const int*   pool_row = (const int*)d_in[42];
  const int*   pool_col = (const int*)d_in[43];
  float* out = (float*)d_out;

  // workspace layout (floats)
  float* ws = (float*)d_ws;
  const size_t TVH  = (size_t)T_ * V_ * 128;
  const size_t TEH  = (size_t)T_ * E_ * 128;
  const size_t TM3H = (size_t)T_ * 10000 * 384;
  float* xv3  = ws;
  float* xe3  = xv3 + TVH;
  float* tA   = xe3 + TEH;
  float* tB   = tA + TEH;
  float* MSG  = tB + TEH;
  float* cntE = MSG + TM3H;
  float* cntV = cntE + E_;
  float* lab  = cntV + V_;
  float* xvf  = lab + T_ * 128;
  float* pcat = xvf + 64 * 128;
  float* p1   = pcat + 96 * 256;
  float* bcat = p1 + 96 * 128;
  float* p2h  = bcat + 16 * 256;

  auto linear = [&](const float* A, const float* W, const float* b, float* C,
                    int N, int K, int relu) {
    wmma_linear_kernel<<<dim3((N + 127) / 128), 256,
                         (size_t)K * 128 * sizeof(float), stream>>>(A, W, b, C, N, K, relu);
  };

  const int NTE = T_ * E_;   // 80000
  const int NTV = T_ * V_;   // 80000
  const size_t convLds = (size_t)384 * 128 * sizeof(float);  // 192 KB < 320 KB/WGP

  // segment counts
  hipMemsetAsync(cntE, 0, E_ * sizeof(float), stream);
  hipMemsetAsync(cntV, 0, V_ * sizeof(float), stream);
  count_kernel<<<(NNZ_ + 255) / 256, 256, 0, stream>>>(inci_row, cntE, NNZ_);
  count_kernel<<<(NNZ_ + 255) / 256, 256, 0, stream>>>(inci_col, cntV, NNZ_);

  // target-node label injection
  label_kernel<<<T_, 128, 0, stream>>>(xv, tar, P[12], P[13], lab);
  init_xv3_kernel<<<T_ * V_, 128, 0, stream>>>(xv, lab, tar, xv3);
  init_xe3_kernel<<<T_ * E_, 128, 0, stream>>>(xe, xe3);

  // initline MLP3 on xe3, initlinv MLP3 on xv3
  linear(xe3, P[0], P[1], tA, NTE, 128, 1);
  linear(tA,  P[2], P[3], tB, NTE, 128, 1);
  linear(tB,  P[4], P[5], xe3, NTE, 128, 0);
  linear(xv3, P[6], P[7], tA, NTV, 128, 1);
  linear(tA,  P[8], P[9], tB, NTV, 128, 1);
  linear(tB,  P[10], P[11], xv3, NTV, 128, 0);

  // conv layers: each layer runs (line, linv) twice
  for (int i = 0; i < 2; ++i) {
    const int lineBase = 14 + i * 8;      // conv{i}_line: W, b, ln_g, ln_b
    const int linvBase = lineBase + 4;    // conv{i}_linv
    for (int rep = 0; rep < 2; ++rep) {
      // edges gather from nodes
      hipMemsetAsync(MSG, 0, (size_t)T_ * E_ * 384 * sizeof(float), stream);
      scatter_kernel<<<T_ * NNZ_, 128, 0, stream>>>(xv3, inci_row, inci_col, MSG,
                                                    NNZ_, V_, E_);
      msg_finalize_kernel<<<(NTE * 128 + 255) / 256, 256, 0, stream>>>(MSG, cntE, E_);
      wmma_conv_kernel<<<dim3((NTE + 127) / 128), 256, convLds, stream>>>(
          MSG, P[lineBase], P[lineBase + 1], P[lineBase + 2], P[lineBase + 3],
          xe3, NTE);

      // nodes gather from edges
      hipMemsetAsync(MSG, 0, (size_t)T_ * V_ * 384 * sizeof(float), stream);
      scatter_kernel<<<T_ * NNZ_, 128, 0, stream>>>(xe3, inci_col, inci_row, MSG,
                                                    NNZ_, E_, V_);
      msg_finalize_kernel<<<(NTV * 128 + 255) / 256, 256, 0, stream>>>(MSG, cntV, V_);
      wmma_conv_kernel<<<dim3((NTV + 127) / 128), 256, convLds, stream>>>(
          MSG, P[linvBase], P[linvBase + 1], P[linvBase + 2], P[linvBase + 3],
          xv3, NTV);
    }
  }

  // pooling head
  xvf_kernel<<<64, 128, 0, stream>>>(xv3, tar, xvf);
  pool_seg_kernel<<<96, 128, 0, stream>>>(xvf, pool_row, pool_col, pcat, 576);
  linear(pcat, P[30], P[31], p1, 96, 256, 1);
  pool_seg_kernel<<<16, 128, 0, stream>>>(p1, batch, nullptr, bcat, 96);
  linear(bcat, P[32], P[33], p2h, 16, 256, 1);
  final_kernel<<<16, 128, 0, stream>>>(p2h, P[34], P[35], out);
}